// PixelContrastLoss_11029476016551
// MI455X (gfx1250) — compile-verified
//
#include <hip/hip_runtime.h>
#include <math.h>

#define A_CNT 256
#define NVIEW 20
#define D_DIM 256
#define NCLS 19
#define N_TOT (A_CNT * NVIEW)   // 5120 rows
#define N_TILES (N_TOT / 16)    // 320 row-tiles of 16

typedef __attribute__((ext_vector_type(2))) float v2f;
typedef __attribute__((ext_vector_type(8))) float v8f;

// ------------------------------------------------------------------
// Pass 1: per-(class,btch) counts.  cntf[c] = total anchors of class c,
// cnt2[2c+b] = anchors of class c with btch==b.
// ------------------------------------------------------------------
__global__ void pcl_prep(const int* __restrict__ labels, const int* __restrict__ btch,
                         float* __restrict__ cntf, int* __restrict__ cnt2g) {
  __shared__ int c2[2 * NCLS];
  const int tid = threadIdx.x;
  if (tid < 2 * NCLS) c2[tid] = 0;
  __syncthreads();
  for (int a = tid; a < A_CNT; a += blockDim.x)
    atomicAdd(&c2[labels[a] * 2 + btch[a]], 1);
  __syncthreads();
  if (tid < NCLS) {
    const int lo = c2[2 * tid], hi = c2[2 * tid + 1];
    cntf[tid] = (float)(lo + hi);
    cnt2g[2 * tid]     = lo;
    cnt2g[2 * tid + 1] = hi;
  } else if (tid < 32) {
    cntf[tid] = 0.f;
  }
}

// ------------------------------------------------------------------
// Pass 2: one wave32 per 16-row tile.  S-tile (16 rows x 32 classes)
// via two V_WMMA_F32_16X16X4_F32 accumulators, K = D = 256.
// Row i = v*A + a  (v = tile/16, a = (tile%16)*16 + m), feat = feats[a,v,:].
// ------------------------------------------------------------------
__global__ void __launch_bounds__(32)
pcl_tile(const float* __restrict__ feats, const float* __restrict__ proto,
         const int* __restrict__ labels, const int* __restrict__ btch,
         const float* __restrict__ cntf, const int* __restrict__ cnt2,
         float* __restrict__ partials) {
  const int t      = blockIdx.x;
  const int lane   = threadIdx.x;        // wave32
  const int laneM  = lane & 15;          // row within tile (A) / class (B)
  const int hi     = lane >> 4;          // 0: K pair {0,1}, 1: K pair {2,3}
  const int koff   = hi * 2;
  const int v      = t >> 4;             // view index
  const int a_base = (t & 15) << 4;      // anchor base

  // A operand: row (a_base+laneM, view v), 2 consecutive K values per lane.
  const float* rowA = feats + ((size_t)(a_base + laneM) * NVIEW + v) * D_DIM + koff;
  // B operands: B[k][c] = proto[c][k]; lane holds class laneM (tile0) / 16+laneM (tile1).
  const float* pB0 = proto + (size_t)laneM * D_DIM + koff;
  const int   c1   = 16 + laneM;
  const float s1   = (c1 < NCLS) ? 1.f : 0.f;          // zero-pad classes 19..31
  const float* pB1 = proto + (size_t)(c1 < NCLS ? c1 : 0) * D_DIM + koff;

  v8f acc0 = {};
  v8f acc1 = {};
#pragma unroll 8
  for (int k = 0; k < D_DIM; k += 4) {
    v2f a  = *(const v2f*)(rowA + k);
    v2f b0 = *(const v2f*)(pB0 + k);
    v2f br = *(const v2f*)(pB1 + k);
    v2f b1;
    b1.x = br.x * s1;
    b1.y = br.y * s1;
    acc0 = __builtin_amdgcn_wmma_f32_16x16x4_f32(false, a, false, b0,
                                                 (short)0, acc0, false, false);
    acc1 = __builtin_amdgcn_wmma_f32_16x16x4_f32(false, a, false, b1,
                                                 (short)0, acc1, false, false);
  }

  // C/D layout: VGPR r holds row r (lanes 0-15, col=lane) and row r+8
  // (lanes 16-31, col=lane-16).  Spill scaled logits S = dot/T to LDS.
  __shared__ float S[16][33];            // +1 pad: no bank conflicts
  const float invT = 10.f;               // 1 / TEMPERATURE
#pragma unroll
  for (int r = 0; r < 8; ++r) {
    const int row = r + hi * 8;
    S[row][laneM]      = acc0[r] * invT;
    S[row][laneM + 16] = acc1[r] * invT;
  }
  __syncthreads();

  // Epilogue: lanes 0..15 handle one row each (all columns of a given class
  // share one logit, so the [N,N] softmax collapses to 19 classes).
  float term = 0.f;
  if (lane < 16) {
    const int a = a_base + lane;
    const int l = labels[a];
    const int b = btch[a];

    float m0 = -INFINITY;                 // row max over present classes
#pragma unroll
    for (int c = 0; c < NCLS; ++c)
      if (cntf[c] > 0.f) m0 = fmaxf(m0, S[lane][c]);

    float neg = 0.f;                      // sum over negatives
#pragma unroll
    for (int c = 0; c < NCLS; ++c)
      if (c != l) neg += cntf[c] * expf(S[lane][c] - m0);
    neg *= (float)NVIEW;

    const float sl = S[lane][l] - m0;
    const float lp = sl - logf(expf(sl) + neg);

    const float pos = (float)(NVIEW * cnt2[l * 2 + (1 - b)]);
    term = (pos * lp) / (pos + 1e-8f);    // mean_log_prob_pos for this row
  }

  // wave reduction (lanes 16..31 contribute 0)
#pragma unroll
  for (int off = 16; off > 0; off >>= 1)
    term += __shfl_xor(term, off, 32);
  if (lane == 0) partials[t] = term;
}

// ------------------------------------------------------------------
// Pass 3: deterministic tree reduction of tile partials -> scalar loss.
// ------------------------------------------------------------------
__global__ void pcl_final(const float* __restrict__ partials, float* __restrict__ out) {
  __shared__ float red[128];
  const int tid = threadIdx.x;
  float s = 0.f;
  for (int i = tid; i < N_TILES; i += 128) s += partials[i];
  red[tid] = s;
  __syncthreads();
  for (int w = 64; w > 0; w >>= 1) {
    if (tid < w) red[tid] += red[tid + w];
    __syncthreads();
  }
  if (tid == 0) {
    float loss = -(0.1f / 0.07f) * red[0] / (float)N_TOT;
    if (isnan(loss)) loss = 0.f;
    out[0] = loss;
  }
}

// ------------------------------------------------------------------
extern "C" void kernel_launch(void* const* d_in, const int* in_sizes, int n_in,
                              void* d_out, int out_size, void* d_ws, size_t ws_size,
                              hipStream_t stream) {
  const float* feats  = (const float*)d_in[0];   // [A, NVIEW, D] f32
  const float* proto  = (const float*)d_in[1];   // [NCLS, D]     f32
  const int*   labels = (const int*)d_in[2];     // [A] i32
  const int*   btch   = (const int*)d_in[3];     // [A] i32

  float* partials = (float*)d_ws;                // 320 floats
  float* cntf     = partials + N_TILES;          // 32 floats
  int*   cnt2     = (int*)(cntf + 32);           // 38 ints

  pcl_prep<<<1, 64, 0, stream>>>(labels, btch, cntf, cnt2);
  pcl_tile<<<N_TILES, 32, 0, stream>>>(feats, proto, labels, btch, cntf, cnt2, partials);
  pcl_final<<<1, 128, 0, stream>>>(partials, (float*)d_out);
}